// Net_83975200571570
// MI455X (gfx1250) — compile-verified
//
#include <hip/hip_runtime.h>
#include <math.h>

#define N_NODES 100000
#define N_EDGES 3200000
#define EP      (N_EDGES + N_NODES)   // edges + self loops
#define NGRAPH  1024
#define FIN     5
#define KPAD1   8                      // layer-1 K padded to multiple of 4
#define HDIM    32
#define LRELU_SLOPE 0.2f

typedef float v2f __attribute__((ext_vector_type(2)));
typedef float v8f __attribute__((ext_vector_type(8)));

// ---------------------------------------------------------------- utilities
__global__ void fill_kernel(float* __restrict__ p, float v, int n) {
    int i = blockIdx.x * blockDim.x + threadIdx.x;
    if (i < n) p[i] = v;
}

// pad x [N x FIN] -> xp [N x KPAD1] with zeros in cols FIN..KPAD1-1
__global__ void pad_x_kernel(const float* __restrict__ x, float* __restrict__ xp) {
    int t = blockIdx.x * blockDim.x + threadIdx.x;
    if (t >= N_NODES * KPAD1) return;
    int n = t >> 3, j = t & (KPAD1 - 1);
    xp[t] = (j < FIN) ? x[n * FIN + j] : 0.0f;
}

__device__ __forceinline__ void atomicMaxF32(float* addr, float v) {
    // signed/unsigned bit trick; addr initialized to -inf
    if (v >= 0.0f) atomicMax((int*)addr, __float_as_int(v));
    else           atomicMin((unsigned int*)addr, __float_as_uint(v));
}

// ---------------------------------------------------------------- WMMA GEMM
// H = X[N x K] @ W[kvalid x 32] (W zero-padded to K rows in LDS), fp32,
// V_WMMA_F32_16X16X4_F32. One wave computes a 16-row x 32-col tile (two
// 16x16 WMMA tiles) accumulating over K/4 chunks. K must be a multiple of 4.
// Block = 256 threads = 8 waves = 128 rows.
template <int K>
__global__ void gemm_wmma_kernel(const float* __restrict__ X,
                                 const float* __restrict__ W,
                                 int kvalid,
                                 float* __restrict__ Hout) {
    __shared__ float sW[K * HDIM];
    // stage weights into LDS once per block, zero-padded past kvalid rows
    for (int i = threadIdx.x; i < K * HDIM; i += 256)
        sW[i] = (i < kvalid * HDIM) ? W[i] : 0.0f;
    __syncthreads();

    const int lane    = threadIdx.x & 31;
    const int wave    = threadIdx.x >> 5;
    const int baseRow = blockIdx.x * 128 + wave * 16;
    const int mrow    = lane & 15;      // A row / B,C col within tile
    const bool hi     = lane >= 16;     // upper half-wave: K+2/K+3 (A,B), rows 8..15 (C)
    // clamp: OOB lanes read row N-1 harmlessly; their results are never stored
    const int row     = min(baseRow + mrow, N_NODES - 1);
    const v2f* arow   = (const v2f*)(X + (size_t)row * K);

    v8f c0 = {};  // cols 0..15
    v8f c1 = {};  // cols 16..31

#pragma unroll
    for (int kc = 0; kc < K / 4; ++kc) {
        const int ka = kc * 4 + (hi ? 2 : 0);
        // A fragment: contiguous pair -> single b64 load
        v2f a = arow[ka >> 1];
        // B fragments from LDS (W row-major K x 32)
        v2f b0, b1;
        b0.x = sW[ka * HDIM + mrow];
        b0.y = sW[(ka + 1) * HDIM + mrow];
        b1.x = sW[ka * HDIM + 16 + mrow];
        b1.y = sW[(ka + 1) * HDIM + 16 + mrow];

        c0 = __builtin_amdgcn_wmma_f32_16x16x4_f32(false, a, false, b0,
                                                   (short)0, c0, false, false);
        c1 = __builtin_amdgcn_wmma_f32_16x16x4_f32(false, a, false, b1,
                                                   (short)0, c1, false, false);
    }

    // C/D layout: VGPR r -> (M=r, N=lane) lanes 0-15, (M=r+8, N=lane-16) lanes 16-31
    const int rbase = baseRow + (hi ? 8 : 0);
    if (baseRow + 15 < N_NODES) {
        // fast path: whole tile in bounds (wave-uniform branch)
#pragma unroll
        for (int r = 0; r < 8; ++r) {
            Hout[(rbase + r) * HDIM + mrow]      = c0[r];
            Hout[(rbase + r) * HDIM + 16 + mrow] = c1[r];
        }
    } else {
#pragma unroll
        for (int r = 0; r < 8; ++r) {
            if (rbase + r < N_NODES) {
                Hout[(rbase + r) * HDIM + mrow]      = c0[r];
                Hout[(rbase + r) * HDIM + 16 + mrow] = c1[r];
            }
        }
    }
}

// ---------------------------------------------------------------- per-node dots
__global__ void node_dots_kernel(const float* __restrict__ Hm,
                                 const float* __restrict__ a_src,
                                 const float* __restrict__ a_dst,
                                 float* __restrict__ asN,
                                 float* __restrict__ adN) {
    int n = blockIdx.x * blockDim.x + threadIdx.x;
    if (n >= N_NODES) return;
    const float4* hp = (const float4*)(Hm + (size_t)n * HDIM);
    float s = 0.0f, d = 0.0f;
#pragma unroll
    for (int q = 0; q < 8; ++q) {
        float4 h = hp[q];
        s += h.x * a_src[q * 4 + 0] + h.y * a_src[q * 4 + 1] +
             h.z * a_src[q * 4 + 2] + h.w * a_src[q * 4 + 3];
        d += h.x * a_dst[q * 4 + 0] + h.y * a_dst[q * 4 + 1] +
             h.z * a_dst[q * 4 + 2] + h.w * a_dst[q * 4 + 3];
    }
    asN[n] = s;
    adN[n] = d;
}

// ---------------------------------------------------------------- edge passes
__global__ void edge_alpha_max_kernel(const int* __restrict__ src,
                                      const int* __restrict__ dst,
                                      const float* __restrict__ asN,
                                      const float* __restrict__ adN,
                                      const float* __restrict__ Hm,
                                      float* __restrict__ alpha,
                                      float* __restrict__ mN) {
    int e = blockIdx.x * blockDim.x + threadIdx.x;
    if (e >= EP) return;
    int s, d;
    if (e < N_EDGES) { s = src[e]; d = dst[e]; }
    else             { s = d = e - N_EDGES; }   // self loop
    // warm L2 with the h[src] row needed by the scatter pass
    __builtin_prefetch(Hm + (size_t)s * HDIM, 0, 1);
    float a = asN[s] + adN[d];
    a = (a > 0.0f) ? a : LRELU_SLOPE * a;       // leaky relu
    alpha[e] = a;
    atomicMaxF32(&mN[d], a);
}

__global__ void edge_exp_kernel(const int* __restrict__ dst,
                                float* __restrict__ alpha,   // in: alpha, out: exp
                                const float* __restrict__ mN,
                                float* __restrict__ denN) {
    int e = blockIdx.x * blockDim.x + threadIdx.x;
    if (e >= EP) return;
    int d = (e < N_EDGES) ? dst[e] : (e - N_EDGES);
    float ex = expf(alpha[e] - mN[d]);
    alpha[e] = ex;
    atomicAdd(&denN[d], ex);
}

__global__ void edge_scatter_kernel(const int* __restrict__ src,
                                    const int* __restrict__ dst,
                                    const float* __restrict__ alpha, // holds exp
                                    const float* __restrict__ denN,
                                    const float* __restrict__ Hm,
                                    float* __restrict__ Out) {
    int e = blockIdx.x * blockDim.x + threadIdx.x;
    if (e >= EP) return;
    int s, d;
    if (e < N_EDGES) { s = src[e]; d = dst[e]; }
    else             { s = d = e - N_EDGES; }
    float w = alpha[e] / denN[d];
    const float4* hp = (const float4*)(Hm + (size_t)s * HDIM);
    float* op = Out + (size_t)d * HDIM;
#pragma unroll
    for (int q = 0; q < 8; ++q) {
        float4 h = hp[q];
        atomicAdd(op + q * 4 + 0, h.x * w);
        atomicAdd(op + q * 4 + 1, h.y * w);
        atomicAdd(op + q * 4 + 2, h.z * w);
        atomicAdd(op + q * 4 + 3, h.w * w);
    }
}

__global__ void bias_add_kernel(float* __restrict__ Out, const float* __restrict__ b) {
    int t = blockIdx.x * blockDim.x + threadIdx.x;
    if (t >= N_NODES * HDIM) return;
    Out[t] += b[t & (HDIM - 1)];
}

// ---------------------------------------------------------------- mean pool
__global__ void pool_accum_kernel(const float* __restrict__ Hm,
                                  const int* __restrict__ batch,
                                  float* __restrict__ sums,
                                  float* __restrict__ cnt) {
    int t = blockIdx.x * blockDim.x + threadIdx.x;
    if (t >= N_NODES * HDIM) return;
    int n = t >> 5;
    int j = t & (HDIM - 1);
    int g = batch[n];
    atomicAdd(&sums[g * HDIM + j], Hm[t]);
    if (j == 0) atomicAdd(&cnt[g], 1.0f);
}

__global__ void pool_div_kernel(const float* __restrict__ sums,
                                const float* __restrict__ cnt,
                                float* __restrict__ out) {
    int t = blockIdx.x * blockDim.x + threadIdx.x;
    if (t >= NGRAPH * HDIM) return;
    float c = cnt[t >> 5];
    out[t] = sums[t] / fmaxf(c, 1.0f);
}

// ---------------------------------------------------------------- launcher
extern "C" void kernel_launch(void* const* d_in, const int* in_sizes, int n_in,
                              void* d_out, int out_size, void* d_ws, size_t ws_size,
                              hipStream_t stream) {
    const float* x     = (const float*)d_in[0];
    const int*   ei    = (const int*)d_in[1];
    const int*   batch = (const int*)d_in[2];
    const float* W[3]  = {(const float*)d_in[3], (const float*)d_in[7],  (const float*)d_in[11]};
    const float* aS[3] = {(const float*)d_in[4], (const float*)d_in[8],  (const float*)d_in[12]};
    const float* aD[3] = {(const float*)d_in[5], (const float*)d_in[9],  (const float*)d_in[13]};
    const float* bb[3] = {(const float*)d_in[6], (const float*)d_in[10], (const float*)d_in[14]};

    const int* src = ei;              // edge_index[0]
    const int* dst = ei + N_EDGES;    // edge_index[1]

    // workspace carve-up
    float* bufA  = (float*)d_ws;                  // [N x 32]  h (post-GEMM features)
    float* bufB  = bufA + (size_t)N_NODES * HDIM; // [N x 32]  layer output / next input
    float* asN   = bufB + (size_t)N_NODES * HDIM; // [N]
    float* adN   = asN + N_NODES;                 // [N]
    float* mN    = adN + N_NODES;                 // [N] segment max
    float* denN  = mN + N_NODES;                  // [N] segment sum
    float* alpha = denN + N_NODES;                // [EP]
    float* sums  = alpha + EP;                    // [G x 32]
    float* cnt   = sums + NGRAPH * HDIM;          // [G]
    float* xpad  = cnt + NGRAPH;                  // [N x 8] zero-padded layer-1 input

    const int B = 256;
    const int gNode  = (N_NODES + B - 1) / B;
    const int gNH    = (N_NODES * HDIM + B - 1) / B;
    const int gNP    = (N_NODES * KPAD1 + B - 1) / B;
    const int gEdge  = (EP + B - 1) / B;
    const int gGemm  = (N_NODES + 127) / 128;
    const int gPoolD = (NGRAPH * HDIM + B - 1) / B;

    // zero-pad layer-1 input once
    pad_x_kernel<<<gNP, B, 0, stream>>>(x, xpad);

    for (int l = 0; l < 3; ++l) {
        // 1) h = x @ W  (WMMA f32 16x16x4, weights staged in LDS)
        if (l == 0)
            gemm_wmma_kernel<KPAD1><<<gGemm, B, 0, stream>>>(xpad, W[0], FIN, bufA);
        else
            gemm_wmma_kernel<HDIM><<<gGemm, B, 0, stream>>>(bufB, W[l], HDIM, bufA);

        // 2) per-node attention dot products
        node_dots_kernel<<<gNode, B, 0, stream>>>(bufA, aS[l], aD[l], asN, adN);

        // 3) init softmax state + output accumulator
        fill_kernel<<<gNode, B, 0, stream>>>(mN, -INFINITY, N_NODES);
        fill_kernel<<<gNode, B, 0, stream>>>(denN, 0.0f, N_NODES);
        fill_kernel<<<gNH, B, 0, stream>>>(bufB, 0.0f, N_NODES * HDIM);

        // 4) scatter-softmax over incoming edges + weighted aggregation
        edge_alpha_max_kernel<<<gEdge, B, 0, stream>>>(src, dst, asN, adN, bufA, alpha, mN);
        edge_exp_kernel<<<gEdge, B, 0, stream>>>(dst, alpha, mN, denN);
        edge_scatter_kernel<<<gEdge, B, 0, stream>>>(src, dst, alpha, denN, bufA, bufB);

        // 5) bias
        bias_add_kernel<<<gNH, B, 0, stream>>>(bufB, bb[l]);
    }

    // global mean pool per graph
    fill_kernel<<<(NGRAPH * HDIM + B - 1) / B, B, 0, stream>>>(sums, 0.0f, NGRAPH * HDIM);
    fill_kernel<<<(NGRAPH + B - 1) / B, B, 0, stream>>>(cnt, 0.0f, NGRAPH);
    pool_accum_kernel<<<gNH, B, 0, stream>>>(bufB, batch, sums, cnt);
    pool_div_kernel<<<gPoolD, B, 0, stream>>>(sums, cnt, (float*)d_out);
}